// GCNLinkPredict_11381663334872
// MI455X (gfx1250) — compile-verified
//
#include <hip/hip_runtime.h>

typedef float v2f __attribute__((ext_vector_type(2)));
typedef float v8f __attribute__((ext_vector_type(8)));

#define N_NODES 100000
#define D 128
#define E_MSG 3200000
#define E_EVAL 500000
#define NROWTILES (N_NODES / 16)   // 6250, exact

// ---------------- degree / normalization ----------------

__global__ void k_init_deg(float* deg) {
    int i = blockIdx.x * 256 + threadIdx.x;
    if (i < N_NODES) deg[i] = 1.0f;   // self-loop contributes 1 to in-degree
}

__global__ void k_count_deg(const long long* __restrict__ dst, float* deg) {
    int e = blockIdx.x * 256 + threadIdx.x;
    if (e < E_MSG) atomicAdd(&deg[(int)dst[e]], 1.0f);
}

__global__ void k_dinv(const float* __restrict__ deg, float* __restrict__ dinv) {
    int i = blockIdx.x * 256 + threadIdx.x;
    if (i < N_NODES) dinv[i] = rsqrtf(deg[i]);  // deg >= 1 always
}

// ---------------- fp32 WMMA GEMM: O[N,128] = act(X)[N,128] @ W[128,128] ------
// One wave per 16-row strip; 8 column tiles; V_WMMA_F32_16X16X4_F32, K=4 steps.
// W staged in LDS pre-swizzled into the exact B-fragment layout (one conflict-
// free ds_load_b64 per tile per K-step). K-loop fully unrolled: accumulators
// chain D->C in SSA form, no rotation movs, no in-loop branches.
// Compile-time fusions:
//   BIAS : A fragments become relu(X + bias)   (layer-2 input)
//   SELF : additionally store O2 = O * dinv[row]^2 (self-loop init, layer 1)

template <bool BIAS, bool SELF>
__global__ __launch_bounds__(256) void k_gemm128(const float* __restrict__ X,
                                                 const float* __restrict__ W,
                                                 float* __restrict__ O,
                                                 const float* __restrict__ bias,
                                                 const float* __restrict__ dinv,
                                                 float* __restrict__ O2) {
    __shared__ float sW[32 * 8 * 64];   // [kb][tile][lane] as float2, 64 KB
    // Pre-swizzle: entry (kb,t,lane) = {W[kb*4+kh][t*16+mr], W[kb*4+kh+2][t*16+mr]}
    for (int idx = threadIdx.x; idx < 32 * 8 * 32; idx += 256) {
        int lane = idx & 31;
        int t    = (idx >> 5) & 7;
        int kb   = idx >> 8;
        int mr   = lane & 15;
        int kh   = lane >> 4;
        v2f v;
        v.x = W[(kb * 4 + kh) * D     + t * 16 + mr];
        v.y = W[(kb * 4 + kh + 2) * D + t * 16 + mr];
        ((v2f*)sW)[idx] = v;
    }
    __syncthreads();

    int wave = (blockIdx.x * 256 + threadIdx.x) >> 5;
    if (wave >= NROWTILES) return;          // wave-uniform: EXEC stays all-ones
    int lane = threadIdx.x & 31;
    int mrow = lane & 15;
    int kh   = lane >> 4;

    const float* xrow = X + (size_t)(wave * 16 + mrow) * D + 2 * kh;
    const v2f*   sB   = (const v2f*)sW + lane;

    v8f acc[8] = {};

#pragma unroll
    for (int kb = 0; kb < 32; ++kb) {
        v2f a = *(const v2f*)(xrow + kb * 4);
        if (BIAS) {
            int c = kb * 4 + 2 * kh;
            a.x = fmaxf(a.x + bias[c],     0.0f);
            a.y = fmaxf(a.y + bias[c + 1], 0.0f);
        }
        const v2f* bp = sB + kb * 256;      // 8 tiles * 32 lanes per kb
#pragma unroll
        for (int t = 0; t < 8; ++t) {
            acc[t] = __builtin_amdgcn_wmma_f32_16x16x4_f32(
                false, a, false, bp[t * 32], (short)0, acc[t], false, false);
        }
    }

    // C/D layout: VGPR i -> row i (lanes 0-15) / row i+8 (lanes 16-31)
    int rowbase = wave * 16 + kh * 8;
#pragma unroll
    for (int i = 0; i < 8; ++i) {
        int row = rowbase + i;
        float* orow = O + (size_t)row * D + mrow;
#pragma unroll
        for (int t = 0; t < 8; ++t)
            orow[t * 16] = acc[t][i];
        if (SELF) {
            float dv = dinv[row]; dv *= dv;
            float* o2row = O2 + (size_t)row * D + mrow;
#pragma unroll
            for (int t = 0; t < 8; ++t)
                o2row[t * 16] = acc[t][i] * dv;
        }
    }
}

// ---------------- aggregation ----------------

// out[i,:] = h[i,:] * dinv[i]^2   (self-loop message, initializes out)
__global__ void k_selfloop_init(const float* __restrict__ h,
                                const float* __restrict__ dinv,
                                float* __restrict__ out) {
    int idx = blockIdx.x * 256 + threadIdx.x;          // one float4 each
    if (idx >= N_NODES * (D / 4)) return;
    int row = idx >> 5;                                // D/4 == 32
    float s = dinv[row]; s *= s;
    float4 v = ((const float4*)h)[idx];
    v.x *= s; v.y *= s; v.z *= s; v.w *= s;
    ((float4*)out)[idx] = v;
}

// one wave per edge: out[dst] += h[src] * dinv[src]*dinv[dst]
__global__ void k_scatter(const float* __restrict__ h,
                          const long long* __restrict__ ei,
                          const float* __restrict__ dinv,
                          float* __restrict__ out) {
    int e = (blockIdx.x * 256 + threadIdx.x) >> 5;
    if (e >= E_MSG) return;
    int lane = threadIdx.x & 31;
    int s = (int)ei[e];
    int d = (int)ei[E_MSG + e];
    float nrm = dinv[s] * dinv[d];
    float4 v = ((const float4*)(h + (size_t)s * D))[lane];
    float* op = out + (size_t)d * D + lane * 4;
    atomicAdd(op + 0, v.x * nrm);
    atomicAdd(op + 1, v.y * nrm);
    atomicAdd(op + 2, v.z * nrm);
    atomicAdd(op + 3, v.w * nrm);
}

// ---------------- link decode: sigmoid(dot(z[s]+b2, z[d]+b2)) ----------------

__global__ void k_decode(const float* __restrict__ z,
                         const float* __restrict__ bias,
                         const long long* __restrict__ ei,
                         float* __restrict__ out, int nE) {
    int e = (blockIdx.x * 256 + threadIdx.x) >> 5;
    if (e >= nE) return;
    int lane = threadIdx.x & 31;
    int s = (int)ei[e];
    int d = (int)ei[nE + e];
    float4 bb = ((const float4*)bias)[lane];
    float4 a = ((const float4*)(z + (size_t)s * D))[lane];
    float4 b = ((const float4*)(z + (size_t)d * D))[lane];
    a.x += bb.x; a.y += bb.y; a.z += bb.z; a.w += bb.w;
    b.x += bb.x; b.y += bb.y; b.z += bb.z; b.w += bb.w;
    float p = a.x * b.x + a.y * b.y + a.z * b.z + a.w * b.w;
#pragma unroll
    for (int off = 16; off > 0; off >>= 1)
        p += __shfl_xor(p, off, 32);
    if (lane == 0) out[e] = 1.0f / (1.0f + __expf(-p));
}

// ---------------- launcher ----------------

extern "C" void kernel_launch(void* const* d_in, const int* in_sizes, int n_in,
                              void* d_out, int out_size, void* d_ws, size_t ws_size,
                              hipStream_t stream) {
    const float*     x    = (const float*)d_in[0];
    const long long* edge = (const long long*)d_in[1];
    const long long* pos  = (const long long*)d_in[2];
    const long long* neg  = (const long long*)d_in[3];
    const float*     W1   = (const float*)d_in[4];
    const float*     b1   = (const float*)d_in[5];
    const float*     W2   = (const float*)d_in[6];
    const float*     b2   = (const float*)d_in[7];
    float* out = (float*)d_out;

    char* ws = (char*)d_ws;
    const size_t featBytes = (size_t)N_NODES * D * sizeof(float);
    float* bufA = (float*)ws;                        // h (GEMM output)
    float* bufB = (float*)(ws + featBytes);          // z (aggregated)
    float* deg  = (float*)(ws + 2 * featBytes);
    float* dinv = deg + N_NODES;

    const int nB  = (N_NODES + 255) / 256;                  // 391
    const int eB  = (E_MSG + 255) / 256;                    // 12500
    const int gB  = (NROWTILES + 7) / 8;                    // 782 (8 waves/block)
    const int slB = (N_NODES * (D / 4) + 255) / 256;        // 12500
    const int scB = (E_MSG * 32 + 255) / 256;               // 400000
    const int dcB = (E_EVAL * 32 + 255) / 256;              // 62500

    // normalization
    k_init_deg<<<nB, 256, 0, stream>>>(deg);
    k_count_deg<<<eB, 256, 0, stream>>>(edge + E_MSG, deg);
    k_dinv<<<nB, 256, 0, stream>>>(deg, dinv);

    // layer 1: h1 = x @ W1 (fused: bufB = h1*dinv^2) ; scatter into bufB
    k_gemm128<false, true><<<gB, 256, 0, stream>>>(x, W1, bufA, nullptr, dinv, bufB);
    k_scatter<<<scB, 256, 0, stream>>>(bufA, edge, dinv, bufB);

    // layer 2: h2 = relu(bufB + b1) @ W2 (bias/relu fused into A loads)
    k_gemm128<true, false><<<gB, 256, 0, stream>>>(bufB, W2, bufA, b1, nullptr, nullptr);
    k_selfloop_init<<<slB, 256, 0, stream>>>(bufA, dinv, bufB);
    k_scatter<<<scB, 256, 0, stream>>>(bufA, edge, dinv, bufB);

    // decode with fused +b2
    k_decode<<<dcB, 256, 0, stream>>>(bufB, b2, pos, out, E_EVAL);
    k_decode<<<dcB, 256, 0, stream>>>(bufB, b2, neg, out + E_EVAL, E_EVAL);
}